// NeuralNetwork_47493748359370
// MI455X (gfx1250) — compile-verified
//
#include <hip/hip_runtime.h>
#include <hip/hip_bf16.h>

// Graph neural scatter-add:
//   sums[dst[e]] += node_values[src[e]] * weights[e]   (20M edges, 4096 bins)
//   out = softmax(sigmoid(sums))                       (4096 outputs)
//
// Memory-bound: 240 MB edge streams / 23.3 TB/s ~= 10.3 us floor.
// CDNA5 strategy:
//   * async global->LDS copy engine (ASYNCcnt) double-buffers the three
//     sequential edge streams into LDS (no VGPR return path, deep queues:
//     8192 waves x 3KB in flight ~= 24 MB to cover HBM latency)
//   * per-block LDS histogram accumulated with ds_add_f32
//   * flush via plain stores to d_ws partials + deterministic column reduce
//     (avoids 2-4M contended global atomics ~ 5-10us of L2 serialization);
//     atomic-flush fallback if ws_size is too small
//   * single-block wave32 softmax finalize

#define NOUT           4096
#define BLOCK          256
#define EDGE_BLOCKS_BW 1024   // partials path: grid sized for ~24 MB in flight
#define EDGE_BLOCKS_AT 256    // atomic path: bound flush atomics to 1M

// ---- gfx1250 async global->LDS copies (builtin + types probe-confirmed) -----
#if defined(__has_builtin)
#if __has_builtin(__builtin_amdgcn_global_load_async_to_lds_b128) && \
    __has_builtin(__builtin_amdgcn_s_wait_asynccnt)
#define USE_ASYNC_LDS 1
#endif
#endif
#ifndef USE_ASYNC_LDS
#define USE_ASYNC_LDS 0
#endif

#if USE_ASYNC_LDS
typedef int v4i __attribute__((vector_size(16)));
typedef __attribute__((address_space(1))) v4i GV4;
typedef __attribute__((address_space(3))) v4i LV4;

// One wave-wide instruction: each lane copies 16B global -> 16B LDS.
__device__ __forceinline__ void async_copy_b128(void* lds_dst, const void* gsrc) {
    __builtin_amdgcn_global_load_async_to_lds_b128(
        (GV4*)(unsigned long long)gsrc,
        (LV4*)(unsigned int)(unsigned long long)lds_dst,  // low 32b == LDS offset
        /*offset=*/0, /*cpol=*/0);
}
#endif

__global__ __launch_bounds__(BLOCK) void zero_sums_kernel(float* __restrict__ sums) {
    sums[blockIdx.x * BLOCK + threadIdx.x] = 0.0f;
}

__global__ __launch_bounds__(BLOCK) void edge_scatter_kernel(
        const float* __restrict__ node_values,
        const float* __restrict__ weights,
        const int*   __restrict__ src,
        const int*   __restrict__ dst,
        float*       __restrict__ sums,       // used when partials == nullptr
        float*       __restrict__ partials,   // [gridDim.x][NOUT] or nullptr
        int n_edges) {
    __shared__ float hist[NOUT];

    #pragma unroll
    for (int i = threadIdx.x; i < NOUT; i += BLOCK) hist[i] = 0.0f;
    __syncthreads();

    const int tidx = threadIdx.x;
    const int n4   = n_edges >> 2;                 // 4-edge packets

    const int4*   src4 = (const int4*)src;
    const int4*   dst4 = (const int4*)dst;
    const float4* w4   = (const float4*)weights;

#if USE_ASYNC_LDS
    // ---- async double-buffered staging of the three edge streams ----------
    __shared__ int4   bs[2][BLOCK];
    __shared__ int4   bd[2][BLOCK];
    __shared__ float4 bw[2][BLOCK];

    const int ntiles = n4 / BLOCK;                 // full 256-packet tiles
    int tile = blockIdx.x;
    if (tile < ntiles) {                           // prologue: stage tile 0
        const size_t p = (size_t)tile * BLOCK + tidx;
        async_copy_b128(&bs[0][tidx], &src4[p]);
        async_copy_b128(&bd[0][tidx], &dst4[p]);
        async_copy_b128(&bw[0][tidx], &w4[p]);
    }
    int buf = 0;
    for (; tile < ntiles; tile += gridDim.x, buf ^= 1) {
        const int nt = tile + gridDim.x;
        if (nt < ntiles) {                         // stage next tile
            const size_t p = (size_t)nt * BLOCK + tidx;
            async_copy_b128(&bs[buf ^ 1][tidx], &src4[p]);
            async_copy_b128(&bd[buf ^ 1][tidx], &dst4[p]);
            async_copy_b128(&bw[buf ^ 1][tidx], &w4[p]);
            __builtin_amdgcn_s_wait_asynccnt(3);   // oldest 3 = current tile
        } else {
            __builtin_amdgcn_s_wait_asynccnt(0);   // drain
        }
        __syncthreads();                           // all waves' copies visible

        const int4   s = bs[buf][tidx];            // ds_load_b128
        const int4   d = bd[buf][tidx];
        const float4 w = bw[buf][tidx];

        const float c0 = node_values[s.x] * w.x;   // L2-resident gathers
        const float c1 = node_values[s.y] * w.y;
        const float c2 = node_values[s.z] * w.z;
        const float c3 = node_values[s.w] * w.w;

        __hip_atomic_fetch_add(&hist[d.x & (NOUT - 1)], c0,
                               __ATOMIC_RELAXED, __HIP_MEMORY_SCOPE_WORKGROUP);
        __hip_atomic_fetch_add(&hist[d.y & (NOUT - 1)], c1,
                               __ATOMIC_RELAXED, __HIP_MEMORY_SCOPE_WORKGROUP);
        __hip_atomic_fetch_add(&hist[d.z & (NOUT - 1)], c2,
                               __ATOMIC_RELAXED, __HIP_MEMORY_SCOPE_WORKGROUP);
        __hip_atomic_fetch_add(&hist[d.w & (NOUT - 1)], c3,
                               __ATOMIC_RELAXED, __HIP_MEMORY_SCOPE_WORKGROUP);

        __syncthreads();                           // buffer safe to overwrite
    }
    const int edges_done = ntiles * BLOCK * 4;
#else
    // ---- fallback: direct B128 grid-stride loads ---------------------------
    const int gtid   = blockIdx.x * BLOCK + tidx;
    const int stride = gridDim.x * BLOCK;
    for (int i = gtid; i < n4; i += stride) {
        __builtin_prefetch(&src4[i + stride], 0, 0);   // global_prefetch_b8
        __builtin_prefetch(&dst4[i + stride], 0, 0);
        __builtin_prefetch(&w4[i + stride], 0, 0);
        const int4   s = src4[i];
        const int4   d = dst4[i];
        const float4 w = w4[i];
        const float c0 = node_values[s.x] * w.x;
        const float c1 = node_values[s.y] * w.y;
        const float c2 = node_values[s.z] * w.z;
        const float c3 = node_values[s.w] * w.w;
        __hip_atomic_fetch_add(&hist[d.x & (NOUT - 1)], c0,
                               __ATOMIC_RELAXED, __HIP_MEMORY_SCOPE_WORKGROUP);
        __hip_atomic_fetch_add(&hist[d.y & (NOUT - 1)], c1,
                               __ATOMIC_RELAXED, __HIP_MEMORY_SCOPE_WORKGROUP);
        __hip_atomic_fetch_add(&hist[d.z & (NOUT - 1)], c2,
                               __ATOMIC_RELAXED, __HIP_MEMORY_SCOPE_WORKGROUP);
        __hip_atomic_fetch_add(&hist[d.w & (NOUT - 1)], c3,
                               __ATOMIC_RELAXED, __HIP_MEMORY_SCOPE_WORKGROUP);
    }
    const int edges_done = n4 * 4;
#endif

    // Tail edges (beyond full tiles / packets): direct scalar path.
    {
        const int gtid   = blockIdx.x * BLOCK + tidx;
        const int stride = gridDim.x * BLOCK;
        for (int i = edges_done + gtid; i < n_edges; i += stride) {
            const float c = node_values[src[i]] * weights[i];
            __hip_atomic_fetch_add(&hist[dst[i] & (NOUT - 1)], c,
                                   __ATOMIC_RELAXED, __HIP_MEMORY_SCOPE_WORKGROUP);
        }
    }
    __syncthreads();

    if (partials != nullptr) {
        // Deterministic flush: coalesced plain stores, full HBM bandwidth.
        float* mine = partials + (size_t)blockIdx.x * NOUT;
        #pragma unroll
        for (int i = tidx; i < NOUT; i += BLOCK) mine[i] = hist[i];
    } else {
        // Fallback flush: hardware global_atomic_add_f32, no return.
        #pragma unroll
        for (int i = tidx; i < NOUT; i += BLOCK) unsafeAtomicAdd(&sums[i], hist[i]);
    }
}

// sums[i] = sum_b partials[b][i]; 32-lane-contiguous reads per step (coalesced).
__global__ __launch_bounds__(BLOCK) void reduce_partials_kernel(
        const float* __restrict__ partials, float* __restrict__ sums, int nblk) {
    const int i = blockIdx.x * BLOCK + threadIdx.x;     // 0..NOUT-1
    const float* p = partials + i;
    float acc = 0.0f;
    int b = 0;
    for (; b + 4 <= nblk; b += 4) {
        acc += p[(size_t)(b + 0) * NOUT];
        acc += p[(size_t)(b + 1) * NOUT];
        acc += p[(size_t)(b + 2) * NOUT];
        acc += p[(size_t)(b + 3) * NOUT];
    }
    for (; b < nblk; ++b) acc += p[(size_t)b * NOUT];
    sums[i] = acc;
}

// One block of 1024 threads (32 wave32s): sigmoid -> exp -> reduce -> normalize.
// sigmoid(s) in (0,1) so exp(sigmoid) in (1,e): no max-subtraction needed.
__global__ __launch_bounds__(1024) void softmax_kernel(const float* __restrict__ sums,
                                                       float* __restrict__ out) {
    __shared__ float wred[32];
    const int tid = threadIdx.x;

    float e[NOUT / 1024];
    float local = 0.0f;
    #pragma unroll
    for (int k = 0; k < NOUT / 1024; ++k) {
        const float s   = sums[tid + k * 1024];
        const float sig = 1.0f / (1.0f + expf(-s));
        const float ex  = expf(sig);
        e[k] = ex;
        local += ex;
    }

    #pragma unroll
    for (int off = 16; off > 0; off >>= 1) local += __shfl_down(local, off, 32);
    if ((tid & 31) == 0) wred[tid >> 5] = local;
    __syncthreads();

    if (tid < 32) {
        float v = wred[tid];
        #pragma unroll
        for (int off = 16; off > 0; off >>= 1) v += __shfl_down(v, off, 32);
        if (tid == 0) wred[0] = v;
    }
    __syncthreads();

    const float inv = 1.0f / wred[0];
    #pragma unroll
    for (int k = 0; k < NOUT / 1024; ++k) out[tid + k * 1024] = e[k] * inv;
}

extern "C" void kernel_launch(void* const* d_in, const int* in_sizes, int n_in,
                              void* d_out, int out_size, void* d_ws, size_t ws_size,
                              hipStream_t stream) {
    (void)n_in; (void)out_size;

    const float* node_values = (const float*)d_in[0];
    const float* weights     = (const float*)d_in[1];
    const int*   src         = (const int*)d_in[2];
    const int*   dst         = (const int*)d_in[3];
    float*       out         = (float*)d_out;
    float*       sums        = (float*)d_ws;   // first 16 KB of scratch
    const int    n_edges     = in_sizes[1];

    const size_t need = (size_t)NOUT * sizeof(float) +
                        (size_t)EDGE_BLOCKS_BW * NOUT * sizeof(float);

    if (ws_size >= need) {
        // Deterministic store+reduce flush (no global atomics).
        float* partials = sums + NOUT;
        edge_scatter_kernel<<<EDGE_BLOCKS_BW, BLOCK, 0, stream>>>(
            node_values, weights, src, dst, sums, partials, n_edges);
        reduce_partials_kernel<<<NOUT / BLOCK, BLOCK, 0, stream>>>(
            partials, sums, EDGE_BLOCKS_BW);
    } else {
        // Atomic-flush fallback (bounded grid), accumulator zeroed every call.
        zero_sums_kernel<<<NOUT / BLOCK, BLOCK, 0, stream>>>(sums);
        edge_scatter_kernel<<<EDGE_BLOCKS_AT, BLOCK, 0, stream>>>(
            node_values, weights, src, dst, sums, nullptr, n_edges);
    }
    softmax_kernel<<<1, 1024, 0, stream>>>(sums, out);
}